// ShiftKernelMaker_4131758539452
// MI455X (gfx1250) — compile-verified
//
#include <hip/hip_runtime.h>

// ---------------------------------------------------------------------------
// Per-slice spatial argmax mask:  out[s, k] = (in[s, k] == max_k in[s, :])
// slices = 128*4096 = 524288, each slice = 7*7 = 49 contiguous f32.
// Memory-bound: ~205 MB total traffic -> ~8.8 us floor @ 23.3 TB/s.
// Data path: async global->LDS DMA (CDNA5 ASYNCcnt path), conflict-free LDS
// reduction, coalesced b128 stores.
// ---------------------------------------------------------------------------

#define KK               49                       // 7*7 elements per slice
#define SLICES_PER_BLOCK 256
#define NTHREADS         256
#define ELEMS_PER_BLOCK  (SLICES_PER_BLOCK * KK)  // 12544 floats = 49 KB LDS
#define VEC4_PER_BLOCK   (ELEMS_PER_BLOCK / 4)    // 3136 float4 per block

#define AS1 __attribute__((address_space(1)))
#define AS3 __attribute__((address_space(3)))

typedef int v4i __attribute__((vector_size(4 * sizeof(int))));

__device__ __forceinline__ void async_load_b128_to_lds(const float* gsrc,
                                                       float* ldst) {
#if defined(__has_builtin) && __has_builtin(__builtin_amdgcn_global_load_async_to_lds_b128)
    __builtin_amdgcn_global_load_async_to_lds_b128(
        (AS1 v4i*)gsrc, (AS3 v4i*)ldst, /*offset=*/0, /*cpol=*/0);
#else
    unsigned lds_off = (unsigned)(unsigned long long)(AS3 void*)ldst;
    asm volatile("global_load_async_to_lds_b128 %0, %1, off"
                 :
                 : "v"(lds_off), "v"(gsrc)
                 : "memory");
#endif
}

__device__ __forceinline__ void async_wait_zero() {
#if defined(__has_builtin) && __has_builtin(__builtin_amdgcn_s_wait_asynccnt)
    __builtin_amdgcn_s_wait_asynccnt(0);
#else
    asm volatile("s_wait_asynccnt 0" ::: "memory");
#endif
}

__global__ void __launch_bounds__(NTHREADS)
shift_mask_kernel(const float* __restrict__ in, float* __restrict__ out) {
    __shared__ float sdata[ELEMS_PER_BLOCK];   // 49 KB staged tile
    __shared__ float smax[SLICES_PER_BLOCK];   // per-slice maxima

    const int tid = threadIdx.x;
    const long long base = (long long)blockIdx.x * ELEMS_PER_BLOCK;
    const float* gin  = in  + base;
    float*       gout = out + base;

    // ---- Phase 1: async DMA global -> LDS, 16 B/lane, fully coalesced ----
    for (int i = tid; i < VEC4_PER_BLOCK; i += NTHREADS) {
        async_load_b128_to_lds(gin + 4 * i, &sdata[4 * i]);
    }
    async_wait_zero();       // this wave's async copies are in LDS
    __syncthreads();         // all waves' tiles visible

    // ---- Phase 2: per-thread slice max. Word stride 49 vs 64 banks,
    //      gcd(49,64)=1 -> all 32 lanes hit distinct banks (conflict-free).
    {
        const int s = tid * KK;
        float m = sdata[s];
#pragma unroll
        for (int k = 1; k < KK; ++k) m = fmaxf(m, sdata[s + k]);
        smax[tid] = m;
    }
    __syncthreads();

    // ---- Phase 3: coalesced b128 mask write (slice = element / 49) ----
    const float4* sdata4 = (const float4*)sdata;
    float4*       gout4  = (float4*)gout;
    for (int i = tid; i < VEC4_PER_BLOCK; i += NTHREADS) {
        float4 d = sdata4[i];
        const unsigned e = 4u * (unsigned)i;
        float4 r;
        r.x = (d.x == smax[(e + 0u) / KK]) ? 1.0f : 0.0f;
        r.y = (d.y == smax[(e + 1u) / KK]) ? 1.0f : 0.0f;
        r.z = (d.z == smax[(e + 2u) / KK]) ? 1.0f : 0.0f;
        r.w = (d.w == smax[(e + 3u) / KK]) ? 1.0f : 0.0f;
        gout4[i] = r;
    }
}

extern "C" void kernel_launch(void* const* d_in, const int* in_sizes, int n_in,
                              void* d_out, int out_size, void* d_ws, size_t ws_size,
                              hipStream_t stream) {
    (void)n_in; (void)d_ws; (void)ws_size; (void)out_size;
    const float* in  = (const float*)d_in[0];
    float*       out = (float*)d_out;

    const int total_elems = in_sizes[0];            // 25,690,112
    const int nslices     = total_elems / KK;       // 524,288
    const int nblocks     = nslices / SLICES_PER_BLOCK;  // 2048 (exact)

    hipLaunchKernelGGL(shift_mask_kernel, dim3(nblocks), dim3(NTHREADS), 0,
                       stream, in, out);
}